// RPN_670014898450
// MI455X (gfx1250) — compile-verified
//
#include <hip/hip_runtime.h>
#include <hip/hip_bf16.h>

typedef float v2f __attribute__((ext_vector_type(2)));
typedef float v8f __attribute__((ext_vector_type(8)));

#define BATCH 8
#define CCH 256
#define HH 64
#define WW 64
#define HW (HH*WW)          // 4096
#define NA 9
#define NANCH (NA*HW)       // 36864
#define NPAD 65536
#define POSTN 1000
#define NMS_THR 0.7f
#define CLIPV 4.135166556742356f

// ---------------- weight reorder -----------------------------------------------
// K index k = tap*256 + ci.  Pair layout so each lane's two A values are adjacent:
// Wp[(k>>1)*512 + co*2 + (k&1)] = conv_w[co][ci][tap]
__global__ void rpn_reorder_w(const float* __restrict__ w, float* __restrict__ Wp) {
    int i = blockIdx.x * 256 + threadIdx.x;
    if (i >= CCH * CCH * 9) return;
    int co  = i / 2304;
    int rem = i % 2304;
    int ci  = rem / 9;
    int tap = rem % 9;
    int k   = tap * 256 + ci;
    Wp[(size_t)(k >> 1) * 512 + co * 2 + (k & 1)] = w[i];
}

// ---------------- 3x3 conv + ReLU via WMMA f32 16x16x4 -------------------------
// grid (W/16, H, B), block 512 (16 waves). Wave w computes co tile [16w,16w+16)
// for 16 consecutive pixels along W.
__global__ void __launch_bounds__(512)
rpn_conv3x3_wmma(const float* __restrict__ x, const float* __restrict__ Wp,
                 const float* __restrict__ bias, float* __restrict__ feats) {
    __shared__ float xs[CCH * 3 * 18];   // [ci][ky 0..2][col 0..17], 55296 B
    const int x0 = blockIdx.x * 16;
    const int y  = blockIdx.y;
    const int b  = blockIdx.z;
    const int tid = threadIdx.x;

    const float* xb = x + (size_t)b * CCH * HW;
    for (int idx = tid; idx < CCH * 54; idx += 512) {
        int ci  = idx / 54;
        int rem = idx % 54;
        int r   = rem / 18;
        int c   = rem % 18;
        int yy = y - 1 + r;
        int xx = x0 - 1 + c;
        float v = 0.0f;
        if (yy >= 0 && yy < HH && xx >= 0 && xx < WW)
            v = xb[(size_t)ci * HW + yy * WW + xx];
        xs[ci * 54 + r * 18 + c] = v;
    }
    __syncthreads();

    const int wave   = tid >> 5;
    const int lane   = tid & 31;
    const int cobase = wave * 16;
    const int l16    = lane & 15;
    const int half   = lane >> 4;        // 0: K = k4+0,k4+1 ; 1: K = k4+2,k4+3
    const int khalf  = half * 2;

    v8f acc = {};
    for (int tap = 0; tap < 9; ++tap) {
        const int ky = tap / 3, kx = tap % 3;
        const float* xt = xs + ky * 18 + (l16 + kx);          // + ci*54 per channel
        // pairIdx for ci4=0:  tap*128 + half ;  advances by 2 per ci4 (=1024 floats)
        const float* wt = Wp + ((size_t)(tap * 128 + half)) * 512 + (cobase + l16) * 2;
#pragma unroll 4
        for (int ci4 = 0; ci4 < 64; ++ci4) {
            const int ci = ci4 * 4 + khalf;
            v2f a, bb;
            a = *(const v2f*)wt;           // A: (W[K=k4+2h][co], W[K=k4+2h+1][co]) as b64
            bb.x = xt[ci * 54];            // B: X[K row = k4+2h][pixel l16]
            bb.y = xt[(ci + 1) * 54];
            acc = __builtin_amdgcn_wmma_f32_16x16x4_f32(
                      false, a, false, bb, (short)0, acc, false, false);
            wt += 1024;
        }
    }

    // C/D layout: VGPR r -> M=r (lanes 0-15) / M=r+8 (lanes 16-31); N = lane%16
    const int px    = x0 + l16;
    const int cooff = half * 8;
    float* fb = feats + (size_t)b * CCH * HW + (size_t)y * WW + px;
#pragma unroll
    for (int r = 0; r < 8; ++r) {
        int co = cobase + cooff + r;
        float v = acc[r] + bias[co];
        fb[(size_t)co * HW] = v > 0.0f ? v : 0.0f;
    }
}

// ---------------- 1x1 heads + anchor decode + clip -----------------------------
// grid (HW/16, B), block 256. Channel mapping: t coord c of anchor a = conv ch c*9+a.
__global__ void __launch_bounds__(256)
rpn_head_decode(const float* __restrict__ feats,
                const float* __restrict__ box_w, const float* __restrict__ box_b,
                const float* __restrict__ obj_w, const float* __restrict__ obj_b,
                float* __restrict__ props, float* __restrict__ scores) {
    __shared__ float fs[CCH * 16];   // [ci][px]
    __shared__ float outs[45 * 16];  // 36 box ch + 9 obj ch
    const int b  = blockIdx.y;
    const int p0 = blockIdx.x * 16;
    const int t  = threadIdx.x;

    const float* fb = feats + (size_t)b * CCH * HW + p0;
    for (int i = t; i < CCH * 16; i += 256) {
        int ci = i >> 4, px = i & 15;
        fs[ci * 16 + px] = fb[(size_t)ci * HW + px];
    }
    __syncthreads();

    for (int o = t; o < 45 * 16; o += 256) {
        int ch = o >> 4, px = o & 15;
        const float* wrow;
        float acc;
        if (ch < 36) { wrow = box_w + ch * 256;        acc = box_b[ch]; }
        else         { wrow = obj_w + (ch - 36) * 256; acc = obj_b[ch - 36]; }
        for (int ci = 0; ci < 256; ++ci)
            acc += wrow[ci] * fs[ci * 16 + px];
        outs[ch * 16 + px] = acc;
    }
    __syncthreads();

    if (t < 144) {
        const int a  = t >> 4, px = t & 15;
        const int p  = p0 + px;
        const int xi = p & 63, yi = p >> 6;
        const int s  = a / 3, r = a % 3;
        const float scale = (s == 0) ? 128.0f : ((s == 1) ? 256.0f : 512.0f);
        const float rat   = (r == 0) ? 0.5f   : ((r == 1) ? 1.0f   : 2.0f);
        const float sq    = sqrtf(rat);
        const float aw0 = scale / sq, ah0 = scale * sq;
        const float cx = (xi + 0.5f) * 16.0f, cy = (yi + 0.5f) * 16.0f;
        const float ax1 = cx - aw0 * 0.5f, ax2 = cx + aw0 * 0.5f;
        const float ay1 = cy - ah0 * 0.5f, ay2 = cy + ah0 * 0.5f;
        const float aw = ax2 - ax1, ah = ay2 - ay1;
        const float acx = ax1 + 0.5f * aw, acy = ay1 + 0.5f * ah;
        const float t0 = outs[(0 * 9 + a) * 16 + px];
        const float t1 = outs[(1 * 9 + a) * 16 + px];
        const float t2 = outs[(2 * 9 + a) * 16 + px];
        const float t3 = outs[(3 * 9 + a) * 16 + px];
        const float cxn = t0 * aw + acx;
        const float cyn = t1 * ah + acy;
        const float wn = expf(fminf(fmaxf(t2, -CLIPV), CLIPV)) * aw;
        const float hn = expf(fminf(fmaxf(t3, -CLIPV), CLIPV)) * ah;
        float bx1 = cxn - 0.5f * wn, by1 = cyn - 0.5f * hn;
        float bx2 = cxn + 0.5f * wn, by2 = cyn + 0.5f * hn;
        bx1 = fminf(fmaxf(bx1, 0.0f), 1024.0f);
        by1 = fminf(fmaxf(by1, 0.0f), 1024.0f);
        bx2 = fminf(fmaxf(bx2, 0.0f), 1024.0f);
        by2 = fminf(fmaxf(by2, 0.0f), 1024.0f);
        const int j = a * HW + p;
        float* pr = props + ((size_t)b * NANCH + j) * 4;
        pr[0] = bx1; pr[1] = by1; pr[2] = bx2; pr[3] = by2;
        scores[(size_t)b * NANCH + j] = outs[(36 + a) * 16 + px];
    }
}

// ---------------- sort setup + global bitonic (stable via (key,idx) order) -----
__global__ void rpn_sort_init(const float* __restrict__ scores,
                              float* __restrict__ keys, int* __restrict__ idxs) {
    const int i = blockIdx.x * 256 + threadIdx.x;   // 0..65535
    const int b = blockIdx.y;
    keys[b * NPAD + i] = (i < NANCH) ? scores[(size_t)b * NANCH + i]
                                     : __builtin_inff();
    idxs[b * NPAD + i] = i;
}

__global__ void rpn_bitonic_pass(float* __restrict__ keys, int* __restrict__ idxs,
                                 int k, int j) {
    const int i = blockIdx.x * 256 + threadIdx.x;   // 0..65535
    const int base = blockIdx.y * NPAD;
    const int ixj = i ^ j;
    if (ixj <= i) return;
    float k1 = keys[base + i], k2 = keys[base + ixj];
    int   i1 = idxs[base + i], i2 = idxs[base + ixj];
    const bool up = ((i & k) == 0);   // ascending direction
    const bool gt = (k1 > k2) || (k1 == k2 && i1 > i2);   // strict total order
    if (gt == up) {
        keys[base + i] = k2; keys[base + ixj] = k1;
        idxs[base + i] = i2; idxs[base + ixj] = i1;
    }
}

// ---------------- top-1000 descending re-sort (tiebreak: ascending position) ---
__global__ void __launch_bounds__(1024)
rpn_top1k(const float* __restrict__ keys, const int* __restrict__ idxs,
          const float* __restrict__ props,
          float* __restrict__ b2, float* __restrict__ s2) {
    __shared__ float ss[1024];
    __shared__ int   pp[1024];
    const int t = threadIdx.x;
    const int b = blockIdx.x;
    ss[t] = (t < POSTN) ? keys[b * NPAD + t] : -__builtin_inff();
    pp[t] = t;
    __syncthreads();
    for (int k = 2; k <= 1024; k <<= 1) {
        for (int j = k >> 1; j > 0; j >>= 1) {
            const int ixj = t ^ j;
            if (ixj > t) {
                const bool dir = ((t & k) == 0);  // true = desired (descending) order
                float sa = ss[t], sb = ss[ixj];
                int   pa = pp[t], pb = pp[ixj];
                // "a precedes b" in desired order: higher score first, tiebreak lower pos
                const bool prec = (sa > sb) || (sa == sb && pa < pb);
                const bool dosw = dir ? !prec : prec;
                if (dosw) { ss[t] = sb; ss[ixj] = sa; pp[t] = pb; pp[ixj] = pa; }
            }
            __syncthreads();
        }
    }
    const int p = pp[t];
    float* ob = b2 + ((size_t)b * 1024 + t) * 4;
    if (p < POSTN) {
        const int jj = idxs[b * NPAD + p];
        const float* pr = props + ((size_t)b * NANCH + jj) * 4;
        ob[0] = pr[0]; ob[1] = pr[1]; ob[2] = pr[2]; ob[3] = pr[3];
        s2[(size_t)b * 1024 + t] = ss[t];
    } else {
        ob[0] = 0.0f; ob[1] = 0.0f; ob[2] = 0.0f; ob[3] = 0.0f;
        s2[(size_t)b * 1024 + t] = 0.0f;
    }
}

// ---------------- greedy NMS + compaction + output -----------------------------
__global__ void __launch_bounds__(1024)
rpn_nms_out(const float* __restrict__ b2, const float* __restrict__ s2,
            float* __restrict__ out) {
    __shared__ float bx[1024 * 4];
    __shared__ float sc[1024];
    __shared__ float ar[1024];
    __shared__ int   keep[1024];
    __shared__ int   scan[1024];
    const int t = threadIdx.x;
    const int b = blockIdx.x;

    const float* ib = b2 + ((size_t)b * 1024 + t) * 4;
    bx[t * 4 + 0] = ib[0]; bx[t * 4 + 1] = ib[1];
    bx[t * 4 + 2] = ib[2]; bx[t * 4 + 3] = ib[3];
    sc[t] = s2[(size_t)b * 1024 + t];
    ar[t] = (ib[2] - ib[0]) * (ib[3] - ib[1]);
    keep[t] = (t < POSTN) ? 1 : 0;
    __syncthreads();

    for (int i = 0; i < POSTN; ++i) {
        __syncthreads();
        if (!keep[i]) continue;             // uniform branch (LDS broadcast)
        if (t > i && t < POSTN && keep[t]) {
            const float ix1 = bx[i * 4 + 0], iy1 = bx[i * 4 + 1];
            const float ix2 = bx[i * 4 + 2], iy2 = bx[i * 4 + 3];
            const float ltx = fmaxf(ix1, bx[t * 4 + 0]);
            const float lty = fmaxf(iy1, bx[t * 4 + 1]);
            const float rbx = fminf(ix2, bx[t * 4 + 2]);
            const float rby = fminf(iy2, bx[t * 4 + 3]);
            const float wv = fmaxf(rbx - ltx, 0.0f);
            const float hv = fmaxf(rby - lty, 0.0f);
            const float inter = wv * hv;
            const float iou = inter / (ar[i] + ar[t] - inter);
            if (iou > NMS_THR) keep[t] = 0;
        }
    }
    __syncthreads();

    // inclusive Hillis-Steele scan over keep
    scan[t] = (t < POSTN) ? keep[t] : 0;
    __syncthreads();
    for (int off = 1; off < 1024; off <<= 1) {
        int v = (t >= off) ? scan[t - off] : 0;
        __syncthreads();
        scan[t] += v;
        __syncthreads();
    }
    const int total = scan[1023];
    float* ob = out + (size_t)b * POSTN * 5;
    if (t < POSTN && keep[t]) {
        const int r = scan[t] - 1;          // exclusive rank
        ob[r * 5 + 0] = bx[t * 4 + 0];
        ob[r * 5 + 1] = bx[t * 4 + 1];
        ob[r * 5 + 2] = bx[t * 4 + 2];
        ob[r * 5 + 3] = bx[t * 4 + 3];
        ob[r * 5 + 4] = sc[t];
    }
    for (int r = total + t; r < POSTN; r += 1024) {
        ob[r * 5 + 0] = 0.0f; ob[r * 5 + 1] = 0.0f; ob[r * 5 + 2] = 0.0f;
        ob[r * 5 + 3] = 0.0f; ob[r * 5 + 4] = 0.0f;
    }
}

// ---------------- launcher -----------------------------------------------------
extern "C" void kernel_launch(void* const* d_in, const int* in_sizes, int n_in,
                              void* d_out, int out_size, void* d_ws, size_t ws_size,
                              hipStream_t stream) {
    const float* x      = (const float*)d_in[0];
    const float* conv_w = (const float*)d_in[1];
    const float* conv_b = (const float*)d_in[2];
    const float* box_w  = (const float*)d_in[3];
    const float* box_b  = (const float*)d_in[4];
    const float* obj_w  = (const float*)d_in[5];
    const float* obj_b  = (const float*)d_in[6];
    float* out = (float*)d_out;

    float* Wp     = (float*)d_ws;                               // 589824
    float* feats  = Wp + 589824;                                // 8388608
    float* props  = feats + (size_t)8388608;                    // 1179648
    float* scores = props + (size_t)1179648;                    // 294912
    float* keys   = scores + (size_t)294912;                    // 524288
    int*   idxs   = (int*)(keys + (size_t)524288);              // 524288
    float* b2     = (float*)(idxs + (size_t)524288);            // 32768
    float* s2     = b2 + (size_t)32768;                         // 8192

    rpn_reorder_w<<<(CCH * CCH * 9 + 255) / 256, 256, 0, stream>>>(conv_w, Wp);

    rpn_conv3x3_wmma<<<dim3(WW / 16, HH, BATCH), 512, 0, stream>>>(x, Wp, conv_b, feats);

    rpn_head_decode<<<dim3(HW / 16, BATCH), 256, 0, stream>>>(
        feats, box_w, box_b, obj_w, obj_b, props, scores);

    rpn_sort_init<<<dim3(NPAD / 256, BATCH), 256, 0, stream>>>(scores, keys, idxs);

    for (int k = 2; k <= NPAD; k <<= 1)
        for (int j = k >> 1; j >= 1; j >>= 1)
            rpn_bitonic_pass<<<dim3(NPAD / 256, BATCH), 256, 0, stream>>>(keys, idxs, k, j);

    rpn_top1k<<<BATCH, 1024, 0, stream>>>(keys, idxs, props, b2, s2);

    rpn_nms_out<<<BATCH, 1024, 0, stream>>>(b2, s2, out);
}